// HyperTurbulentGenerator_15126874817260
// MI455X (gfx1250) — compile-verified
//
#include <hip/hip_runtime.h>
#include <hip/hip_bf16.h>

// Problem constants (match reference)
#define BB  16   // batch
#define SS  512  // time steps
#define DD  64   // state dim
#define NN  8    // basis count
#define KT  32   // Taylor terms: ||A t|| <~ 1, 32 terms => < fp32 eps

typedef __attribute__((ext_vector_type(2))) float v2f;
typedef __attribute__((ext_vector_type(8))) float v8f;

// ---------------------------------------------------------------------------
// Kernel 1: per-batch A = alpha*(K - K^T) - beta*(R^T R)
// gram = R^T R done with fp32 WMMA 16x16x4 (exact fp32 GEMM path on CDNA5)
// ---------------------------------------------------------------------------
__global__ __launch_bounds__(256)
void build_A_kernel(const float* __restrict__ kc, const float* __restrict__ rc,
                    const float* __restrict__ alpha, const float* __restrict__ beta,
                    const float* __restrict__ Kb, const float* __restrict__ Rb,
                    float* __restrict__ dA)
{
    __shared__ float Ks[DD * DD];
    __shared__ float Rs[DD * DD];

    const int b   = blockIdx.x;
    const int tid = threadIdx.x;

    float kcoef[NN], rcoef[NN];
#pragma unroll
    for (int n = 0; n < NN; ++n) {
        kcoef[n] = kc[b * NN + n];
        rcoef[n] = rc[b * NN + n];
    }
    const float a  = alpha[b];
    const float be = beta[b];

    // K_sum / R_sum = sum_n coeff[n] * bases[n]  (4096 elems, 256 threads)
    for (int idx = tid; idx < DD * DD; idx += 256) {
        float ks = 0.f, rs = 0.f;
#pragma unroll
        for (int n = 0; n < NN; ++n) {
            ks = fmaf(kcoef[n], Kb[n * DD * DD + idx], ks);
            rs = fmaf(rcoef[n], Rb[n * DD * DD + idx], rs);
        }
        Ks[idx] = ks;
        Rs[idx] = rs;
    }
    __syncthreads();

    const int wave = tid >> 5;      // 8 waves (wave32)
    const int lane = tid & 31;
    const int half = lane >> 4;     // 0: lanes 0-15, 1: lanes 16-31
    const int l15  = lane & 15;

    // gram tiles: 4x4 grid of 16x16 tiles, 2 tiles per wave.
    // gram[i,j] = sum_k R[k,i]*R[k,j]  => A-frag[M=i][K]=Rs[K*64+i], B-frag[K][N=j]=Rs[K*64+j]
    for (int t = wave; t < 16; t += 8) {
        const int ti = t >> 2;
        const int tj = t & 3;
        v8f c = {};
#pragma unroll
        for (int kk = 0; kk < DD; kk += 4) {
            const int krow = kk + half * 2;        // ISA layout: upper half-wave holds K=2,3
            v2f af, bf;
            af.x = Rs[(krow + 0) * DD + ti * 16 + l15];
            af.y = Rs[(krow + 1) * DD + ti * 16 + l15];
            bf.x = Rs[(krow + 0) * DD + tj * 16 + l15];
            bf.y = Rs[(krow + 1) * DD + tj * 16 + l15];
            c = __builtin_amdgcn_wmma_f32_16x16x4_f32(false, af, false, bf,
                                                      (short)0, c, false, false);
        }
        // C layout: VGPR r -> M = r + 8*half, N = l15
#pragma unroll
        for (int r = 0; r < 8; ++r) {
            const int i = ti * 16 + r + 8 * half;
            const int j = tj * 16 + l15;
            const float val = a * (Ks[i * DD + j] - Ks[j * DD + i]) - be * c[r];
            dA[b * DD * DD + i * DD + j] = val;
        }
    }
}

// ---------------------------------------------------------------------------
// Kernel 2: Vandermonde table T[k][s] = t_s^k  (32 x 512)
// ---------------------------------------------------------------------------
__global__ void vander_kernel(const float* __restrict__ ts, float* __restrict__ T)
{
    const int s = blockIdx.x * blockDim.x + threadIdx.x;
    if (s < SS) {
        const float t = ts[s];
        float p = 1.f;
#pragma unroll
        for (int k = 0; k < KT; ++k) {
            T[k * SS + s] = p;
            p *= t;
        }
    }
}

// ---------------------------------------------------------------------------
// Kernel 3: Krylov columns  u_k = A^k z / k!   ->  U[o][k]  (64 x 32 per (b,v))
// One block per (b, vector), 64 threads (thread o owns output row o).
// A stored transposed in LDS so the per-thread dot is bank-conflict-free.
// ---------------------------------------------------------------------------
__global__ __launch_bounds__(64)
void krylov_kernel(const float* __restrict__ zr, const float* __restrict__ zi,
                   const float* __restrict__ dA, float* __restrict__ dU)
{
    __shared__ float At[DD * DD];   // At[j*64 + o] = A[o][j]
    __shared__ float vb[2][DD];

    const int b = blockIdx.x >> 1;
    const int v = blockIdx.x & 1;
    const int o = threadIdx.x;

    for (int idx = o; idx < DD * DD; idx += 64) {
        const int oo = idx >> 6;
        const int jj = idx & 63;
        At[jj * DD + oo] = dA[b * DD * DD + idx];
    }

    const float* z = v ? zi : zr;
    const float x = z[b * DD + o];
    vb[0][o] = x;

    float* Urow = dU + ((size_t)(b * 2 + v) * DD + o) * KT;
    Urow[0] = x;
    __syncthreads();

    int cur = 0;
    for (int k = 1; k < KT; ++k) {
        float acc = 0.f;
#pragma unroll 8
        for (int j = 0; j < DD; ++j)
            acc = fmaf(At[j * DD + o], vb[cur][j], acc);
        acc *= (1.0f / (float)k);          // fold 1/k! into the column
        vb[cur ^ 1][o] = acc;
        Urow[k] = acc;
        __syncthreads();
        cur ^= 1;
    }
}

// ---------------------------------------------------------------------------
// Kernel 4: out[v][b][s][o] = sum_k U[o][k] * T[k][s]
// Per (b,v): C(64x512) = U(64x32) x T(32x512), fp32 WMMA 16x16x4.
// 4 M-tiles x 32 N-tiles = 128 tiles, 8 k-steps each; 8 waves per block.
// ---------------------------------------------------------------------------
__global__ __launch_bounds__(256)
void expand_kernel(const float* __restrict__ dU, const float* __restrict__ dT,
                   float* __restrict__ out)
{
    const int bv   = blockIdx.x;        // 0..31
    const int b    = bv >> 1;
    const int v    = bv & 1;
    const int wave = threadIdx.x >> 5;
    const int lane = threadIdx.x & 31;
    const int half = lane >> 4;
    const int l15  = lane & 15;

    const float* U = dU + (size_t)bv * DD * KT;           // U[o*KT + k]
    float* ob = out + ((size_t)(v * BB + b)) * SS * DD;   // out[s*64 + o]

    for (int t = wave; t < 128; t += 8) {
        const int ti = t >> 5;          // 0..3   (M tile)
        const int tn = t & 31;          // 0..31  (N tile)
        const int orow = ti * 16 + l15;
        const int scol = tn * 16 + l15;
        v8f c = {};
#pragma unroll
        for (int kk = 0; kk < KT; kk += 4) {
            const int kb = kk + half * 2;
            v2f af, bf;
            af.x = U[orow * KT + kb];
            af.y = U[orow * KT + kb + 1];
            bf.x = dT[(kb + 0) * SS + scol];
            bf.y = dT[(kb + 1) * SS + scol];
            c = __builtin_amdgcn_wmma_f32_16x16x4_f32(false, af, false, bf,
                                                      (short)0, c, false, false);
        }
#pragma unroll
        for (int r = 0; r < 8; ++r) {
            const int o = ti * 16 + r + 8 * half;
            const int s = tn * 16 + l15;
            ob[(size_t)s * DD + o] = c[r];
        }
    }
}

// ---------------------------------------------------------------------------
// Launch
// ---------------------------------------------------------------------------
extern "C" void kernel_launch(void* const* d_in, const int* in_sizes, int n_in,
                              void* d_out, int out_size, void* d_ws, size_t ws_size,
                              hipStream_t stream)
{
    const float* z0_real    = (const float*)d_in[0];  // (B,D)
    const float* z0_imag    = (const float*)d_in[1];  // (B,D)
    const float* time_steps = (const float*)d_in[2];  // (S,)
    const float* k_coeffs   = (const float*)d_in[3];  // (B,N)
    const float* r_coeffs   = (const float*)d_in[4];  // (B,N)
    const float* alpha      = (const float*)d_in[5];  // (B,1)
    const float* beta       = (const float*)d_in[6];  // (B,1)
    const float* K_bases    = (const float*)d_in[7];  // (N,D,D)
    const float* R_bases    = (const float*)d_in[8];  // (N,D,D)
    float*       out        = (float*)d_out;          // (2,B,S,D)

    // Workspace partition (all fp32):
    //   dA: B*64*64      = 262144 B
    //   dU: B*2*64*KT    = 262144 B
    //   dT: KT*S         =  65536 B
    char* ws = (char*)d_ws;
    float* dA = (float*)(ws);
    float* dU = (float*)(ws + 256 * 1024);
    float* dT = (float*)(ws + 512 * 1024);

    build_A_kernel<<<BB, 256, 0, stream>>>(k_coeffs, r_coeffs, alpha, beta,
                                           K_bases, R_bases, dA);
    vander_kernel<<<(SS + 255) / 256, 256, 0, stream>>>(time_steps, dT);
    krylov_kernel<<<BB * 2, 64, 0, stream>>>(z0_real, z0_imag, dA, dU);
    expand_kernel<<<BB * 2, 256, 0, stream>>>(dU, dT, out);
}